// SocialConvLSTM_63608465653761
// MI455X (gfx1250) — compile-verified
//
#include <hip/hip_runtime.h>
#include <math.h>

// ---------------------------------------------------------------------------
// SocialConvLSTM for MI455X (gfx1250), fp32 end-to-end using
// V_WMMA_F32_16X16X4_F32 (exact fp32 matmul on the matrix pipes).
// ---------------------------------------------------------------------------

typedef float v2f __attribute__((ext_vector_type(2)));
typedef float v8f __attribute__((ext_vector_type(8)));

#define OBS_LEN 16
#define PRED_LEN 25
#define NN 192
#define PP (NN * NN)        // 36864
#define ENC 64
#define DEC 128
#define EMB 32
#define DYN 32
#define C1 64
#define C2 16
#define SOCD 144
#define ENCIN (DYN + SOCD)  // 176

__device__ __forceinline__ float sigmoidf_(float x) { return 1.0f / (1.0f + __expf(-x)); }
__device__ __forceinline__ float lrelu_(float x) { return x >= 0.f ? x : 0.1f * x; }

__device__ __forceinline__ v8f zero8() {
    v8f z = {0.f, 0.f, 0.f, 0.f, 0.f, 0.f, 0.f, 0.f};
    return z;
}

// D = A(16x4 f32) * B(4x16 f32) + C(16x16 f32), wave32.
// Arg pattern per probe doc: (neg_a, A, neg_b, B, c_mod, C, reuse_a, reuse_b)
__device__ __forceinline__ v8f wmma4(v2f a, v2f b, v8f c) {
    return __builtin_amdgcn_wmma_f32_16x16x4_f32(false, a, false, b, (short)0, c, false, false);
}

// ---------------------------------------------------------------------------
// Encoder LSTM step (shared weights for ego + neighbor encoders).
// Block = 256 threads = 8 waves; wave handles a 16-row tile.
// z = lrelu(x@Wip^T+bip) @ Wih^T + H[idx] @ Whh^T + (bih+bhh)
// Fused B matrix: rows k<32 -> Wih, k in [32,96) -> Whh, staged in LDS in
// WMMA fragment layout:  float2 at [(ks*16+nt)*32 + lane].
// ---------------------------------------------------------------------------
#define ENC_BFRAG_FLOATS 24576  // 24 ksteps * 16 ntiles * 32 lanes * 2
#define ENC_SMEM_FLOATS (ENC_BFRAG_FLOATS + 64 + 32 + 256 + 128)

__global__ __launch_bounds__(256) void lstm_enc_step(
    const float* __restrict__ frames,  // [n,2], this timestep
    const int* __restrict__ idxs,      // [n],   this timestep
    const float* __restrict__ Hin, const float* __restrict__ Cin,
    float* __restrict__ Hout, float* __restrict__ Cout,
    const float* __restrict__ Wip, const float* __restrict__ bip,
    const float* __restrict__ Wih, const float* __restrict__ Whh,
    const float* __restrict__ bih, const float* __restrict__ bhh, int n)
{
    extern __shared__ float sm[];
    float* bF   = sm;                   // 24576 floats (96KB)  B fragments
    float* wipS = bF + ENC_BFRAG_FLOATS;// 64
    float* bipS = wipS + 64;            // 32
    float* bsum = bipS + 32;            // 256
    int*   idxS = (int*)(bsum + 256);   // 128 ints (8 waves * 16 rows)

    const int tid = threadIdx.x;
    if (tid == 0) { __builtin_prefetch(Wih, 0, 1); __builtin_prefetch(Whh, 0, 1); }

    // Stage fused B fragments: value = W[n_col][k], n_col from (nt,lane), k from (ks,comp,lanehalf)
    for (int f = tid; f < ENC_BFRAG_FLOATS; f += 256) {
        int comp = f & 1;
        int q = f >> 1;
        int lane = q & 31, nt = (q >> 5) & 15, ks = q >> 9;
        int k = ks * 4 + comp + ((lane & 16) ? 2 : 0);
        int ncol = nt * 16 + (lane & 15);
        bF[f] = (k < EMB) ? Wih[ncol * EMB + k] : Whh[ncol * ENC + (k - EMB)];
    }
    for (int f = tid; f < 64; f += 256) wipS[f] = Wip[f];
    for (int f = tid; f < 32; f += 256) bipS[f] = bip[f];
    for (int f = tid; f < 256; f += 256) bsum[f] = bih[f] + bhh[f];

    const int wave = tid >> 5, lane = tid & 31;
    const int m16 = lane & 15, hi = lane >> 4;
    const long row0 = (long)(blockIdx.x * 8 + wave) * 16;
    if (lane < 16) {
        long fr = row0 + m16;
        idxS[wave * 16 + m16] = (fr < n) ? idxs[fr] : 0;
    }
    __syncthreads();
    if (row0 + 15 >= n) return;  // wave-uniform

    const int sIdx = idxS[wave * 16 + m16];  // this lane's row -> state slot

    // A fragments for the embedded input: emb = lrelu(x @ Wip^T + bip), K = 0..31
    v2f x = *(const v2f*)(frames + (row0 + m16) * 2);
    v2f a_emb[8];
#pragma unroll
    for (int ks = 0; ks < 8; ++ks) {
        int e0 = ks * 4 + 2 * hi;
        float v0 = lrelu_(x.x * wipS[e0 * 2 + 0] + x.y * wipS[e0 * 2 + 1] + bipS[e0]);
        float v1 = lrelu_(x.x * wipS[e0 * 2 + 2] + x.y * wipS[e0 * 2 + 3] + bipS[e0 + 1]);
        v2f a; a.x = v0; a.y = v1;
        a_emb[ks] = a;
    }
    // A fragments for the hidden state: K = 32..95 -> H[sIdx][0..63]
    v2f a_h[16];
    const float* hrow = Hin + (long)sIdx * ENC;
#pragma unroll
    for (int ks = 0; ks < 16; ++ks)
        a_h[ks] = *(const v2f*)(hrow + ks * 4 + 2 * hi);

    const v2f* bQ = (const v2f*)bF;

    for (int g = 0; g < 4; ++g) {  // gate-group: ntiles {g, g+4, g+8, g+12} = i,f,g,o
        v8f ai = zero8(), af = zero8(), ag = zero8(), ao = zero8();
#pragma unroll
        for (int ks = 0; ks < 24; ++ks) {
            v2f a = (ks < 8) ? a_emb[ks] : a_h[ks - 8];
            ai = wmma4(a, bQ[ks * 512 + (g     ) * 32 + lane], ai);
            af = wmma4(a, bQ[ks * 512 + (g +  4) * 32 + lane], af);
            ag = wmma4(a, bQ[ks * 512 + (g +  8) * 32 + lane], ag);
            ao = wmma4(a, bQ[ks * 512 + (g + 12) * 32 + lane], ao);
        }
        const float bi_s = bsum[g * 16 + m16];
        const float bf_s = bsum[64 + g * 16 + m16];
        const float bg_s = bsum[128 + g * 16 + m16];
        const float bo_s = bsum[192 + g * 16 + m16];
        const int col = g * 16 + m16;  // hidden-unit column in [0,64)
#pragma unroll
        for (int j = 0; j < 8; ++j) {
            int sRow = idxS[wave * 16 + j + hi * 8];
            float iv = sigmoidf_(ai[j] + bi_s);
            float fv = sigmoidf_(af[j] + bf_s);
            float gv = tanhf(ag[j] + bg_s);
            float ov = sigmoidf_(ao[j] + bo_s);
            float cold = Cin[(long)sRow * ENC + col];
            float cn = fv * cold + iv * gv;
            Hout[(long)sRow * ENC + col] = ov * tanhf(cn);
            Cout[(long)sRow * ENC + col] = cn;
        }
    }
}

// ---------------------------------------------------------------------------
// traj_enc = lrelu(Hs[idx15] @ Wdyn^T + bdyn) -> enc[:, 0:32]
// ---------------------------------------------------------------------------
__global__ void traj_enc_kernel(const float* __restrict__ Hs, const int* __restrict__ idx15,
                                const float* __restrict__ Wdyn, const float* __restrict__ bdyn,
                                float* __restrict__ enc)
{
    int t = blockIdx.x * blockDim.x + threadIdx.x;
    if (t >= NN * DYN) return;
    int b = t / DYN, d = t % DYN;
    const float* h = Hs + (long)idx15[b] * ENC;
    float acc = bdyn[d];
    for (int k = 0; k < ENC; ++k) acc += h[k] * Wdyn[d * ENC + k];
    enc[b * ENCIN + d] = lrelu_(acc);
}

// sgrid[b][e][gw][gh] = mask[b][gh][gw][e] ? Hn[masks_idxs[(b*8+gh)*8+gw]][e] : 0
__global__ void soc_gather(const float* __restrict__ Hn, const int* __restrict__ midx,
                           const unsigned char* __restrict__ masks, float* __restrict__ sgrid)
{
    int t = blockIdx.x * blockDim.x + threadIdx.x;
    if (t >= NN * ENC * 64) return;
    int gh = t & 7, gw = (t >> 3) & 7, e = (t >> 6) & 63, b = t >> 12;
    int cell = (b * 8 + gh) * 8 + gw;
    sgrid[t] = masks[cell * 64 + e] ? Hn[(long)midx[cell] * ENC + e] : 0.f;
}

__global__ void conv1_k(const float* __restrict__ in, const float* __restrict__ W,
                        const float* __restrict__ bias, float* __restrict__ out)
{
    int t = blockIdx.x * blockDim.x + threadIdx.x;
    if (t >= NN * C1 * 36) return;
    int xv = t % 6, yv = (t / 6) % 6, c = (t / 36) % C1, b = t / (36 * C1);
    float acc = bias[c];
    const float* inB = in + (long)b * ENC * 64;
    const float* wC = W + (long)c * ENC * 9;
    for (int ic = 0; ic < ENC; ++ic) {
        const float* ip = inB + ic * 64 + yv * 8 + xv;
        const float* wp = wC + ic * 9;
#pragma unroll
        for (int ky = 0; ky < 3; ++ky)
#pragma unroll
            for (int kx = 0; kx < 3; ++kx)
                acc += ip[ky * 8 + kx] * wp[ky * 3 + kx];
    }
    out[t] = lrelu_(acc);
}

__global__ void conv2_k(const float* __restrict__ in, const float* __restrict__ W,
                        const float* __restrict__ bias, float* __restrict__ out)
{
    int t = blockIdx.x * blockDim.x + threadIdx.x;
    if (t >= NN * C2 * 16) return;
    int xv = t % 4, yv = (t / 4) % 4, c = (t / 16) % C2, b = t / (16 * C2);
    float acc = bias[c];
    for (int ic = 0; ic < C1; ++ic) {
        const float* ip = in + ((long)(b * C1 + ic) * 6 + yv) * 6 + xv;
        const float* wp = W + (long)(c * C1 + ic) * 9;
#pragma unroll
        for (int ky = 0; ky < 3; ++ky)
#pragma unroll
            for (int kx = 0; kx < 3; ++kx)
                acc += ip[ky * 6 + kx] * wp[ky * 3 + kx];
    }
    out[t] = lrelu_(acc);
}

// 2x2 stride-1 max pool of [B,16,4,4] -> enc[:, 32 + c*9 + y*3 + x]
__global__ void pool_k(const float* __restrict__ in, float* __restrict__ enc)
{
    int t = blockIdx.x * blockDim.x + threadIdx.x;
    if (t >= NN * C2 * 9) return;
    int xv = t % 3, yv = (t / 3) % 3, c = (t / 9) % C2, b = t / (9 * C2);
    const float* p = in + ((long)(b * C2 + c) * 4 + yv) * 4 + xv;
    float m = fmaxf(fmaxf(p[0], p[1]), fmaxf(p[4], p[5]));
    enc[b * ENCIN + DYN + c * 9 + yv * 3 + xv] = m;
}

// xpart[b][nI] = enc[b] . Wih_d[nI] + bih_d[nI] + bhh_d[nI]  (constant across scan)
__global__ void xpart_k(const float* __restrict__ enc, const float* __restrict__ Wih_d,
                        const float* __restrict__ bih_d, const float* __restrict__ bhh_d,
                        float* __restrict__ xp)
{
    int t = blockIdx.x * blockDim.x + threadIdx.x;
    if (t >= NN * 512) return;
    int nI = t % 512, b = t / 512;
    float acc = bih_d[nI] + bhh_d[nI];
    const float* e = enc + b * ENCIN;
    const float* w = Wih_d + (long)nI * ENCIN;
    for (int k = 0; k < ENCIN; ++k) acc += e[k] * w[k];
    xp[t] = acc;
}

// ---------------------------------------------------------------------------
// Decoder LSTM step: z = xpart + h @ Whh_d^T, DEC=128. Block = 128 thr = 4 waves;
// block covers one gate-group g and 4 row-tiles. Whh fragments (32KB) in LDS.
// ---------------------------------------------------------------------------
__global__ __launch_bounds__(128) void lstm_dec_step(
    const float* __restrict__ Hprev, const float* __restrict__ Cprev,
    float* __restrict__ Hout, float* __restrict__ Cout,
    const float* __restrict__ Whh, const float* __restrict__ xp)
{
    __shared__ __attribute__((aligned(16))) float bF[8192];  // 32 ks * 4 q * 32 lanes * 2
    const int tid = threadIdx.x;
    const int g = blockIdx.x & 7;
    const int blockRow = blockIdx.x >> 3;

    for (int f = tid; f < 8192; f += 128) {
        int comp = f & 1;
        int q = f >> 1;
        int lane = q & 31, qq = (q >> 5) & 3, ks = q >> 7;
        int k = ks * 4 + comp + ((lane & 16) ? 2 : 0);
        int ncol = qq * 128 + g * 16 + (lane & 15);
        bF[f] = Whh[(long)ncol * DEC + k];
    }
    __syncthreads();

    const int wave = tid >> 5, lane = tid & 31;
    const int m16 = lane & 15, hi = lane >> 4;
    const int row0 = (blockRow * 4 + wave) * 16;

    v2f a_h[32];
    const float* hrow = Hprev + (long)(row0 + m16) * DEC;
#pragma unroll
    for (int ks = 0; ks < 32; ++ks)
        a_h[ks] = *(const v2f*)(hrow + ks * 4 + 2 * hi);

    const v2f* bQ = (const v2f*)bF;
    v8f ai = zero8(), af = zero8(), ag = zero8(), ao = zero8();
#pragma unroll 8
    for (int ks = 0; ks < 32; ++ks) {
        v2f a = a_h[ks];
        ai = wmma4(a, bQ[ks * 128 + 0 * 32 + lane], ai);
        af = wmma4(a, bQ[ks * 128 + 1 * 32 + lane], af);
        ag = wmma4(a, bQ[ks * 128 + 2 * 32 + lane], ag);
        ao = wmma4(a, bQ[ks * 128 + 3 * 32 + lane], ao);
    }

    const int col = g * 16 + m16;  // hidden-unit column in [0,128)
#pragma unroll
    for (int j = 0; j < 8; ++j) {
        int row = row0 + j + hi * 8;
        float iv = sigmoidf_(ai[j] + xp[(long)row * 512 + col]);
        float fv = sigmoidf_(af[j] + xp[(long)row * 512 + 128 + col]);
        float gv = tanhf(ag[j] + xp[(long)row * 512 + 256 + col]);
        float ov = sigmoidf_(ao[j] + xp[(long)row * 512 + 384 + col]);
        float cold = Cprev[(long)row * DEC + col];
        float cn = fv * cold + iv * gv;
        Hout[(long)row * DEC + col] = ov * tanhf(cn);
        Cout[(long)row * DEC + col] = cn;
    }
}

// out[t][b][o]: mu_x, mu_y, exp(sx), exp(sy), tanh(corr)
__global__ void out_k(const float* __restrict__ hs, const float* __restrict__ Wop,
                      const float* __restrict__ bop, float* __restrict__ out)
{
    int t = blockIdx.x * blockDim.x + threadIdx.x;
    if (t >= PRED_LEN * NN * 5) return;
    int o = t % 5, b = (t / 5) % NN, step = t / (5 * NN);
    const float* h = hs + ((long)step * NN + b) * DEC;
    float acc = bop[o];
    for (int k = 0; k < DEC; ++k) acc += h[k] * Wop[o * DEC + k];
    if (o == 2 || o == 3) acc = __expf(acc);
    else if (o == 4) acc = tanhf(acc);
    out[t] = acc;
}

// ---------------------------------------------------------------------------
extern "C" void kernel_launch(void* const* d_in, const int* in_sizes, int n_in,
                              void* d_out, int out_size, void* d_ws, size_t ws_size,
                              hipStream_t stream)
{
    const float* obs_traj  = (const float*)d_in[0];
    const float* obs_ngbrs = (const float*)d_in[1];
    const unsigned char* masks = (const unsigned char*)d_in[2];  // jnp bool -> 1 byte
    const int* masks_idxs = (const int*)d_in[3];
    const int* obs_idxs   = (const int*)d_in[4];
    const int* ngbr_idxs  = (const int*)d_in[5];
    // d_in[6] = num_nodes (compile-time NN)
    const float* Wip = (const float*)d_in[7];
    const float* bip = (const float*)d_in[8];
    const float* Wih_e = (const float*)d_in[9];
    const float* Whh_e = (const float*)d_in[10];
    const float* bih_e = (const float*)d_in[11];
    const float* bhh_e = (const float*)d_in[12];
    const float* Wdyn = (const float*)d_in[13];
    const float* bdyn = (const float*)d_in[14];
    const float* Wc1 = (const float*)d_in[15];
    const float* bc1 = (const float*)d_in[16];
    const float* Wc2 = (const float*)d_in[17];
    const float* bc2 = (const float*)d_in[18];
    const float* Wih_d = (const float*)d_in[19];
    const float* Whh_d = (const float*)d_in[20];
    const float* bih_d = (const float*)d_in[21];
    const float* bhh_d = (const float*)d_in[22];
    const float* Wop = (const float*)d_in[23];
    const float* bop = (const float*)d_in[24];
    (void)in_sizes; (void)n_in; (void)out_size; (void)ws_size;

    char* ws = (char*)d_ws;
    size_t off = 0;
    auto alloc = [&](size_t bytes) -> float* {
        float* p = (float*)(ws + off);
        off += (bytes + 255) & ~(size_t)255;
        return p;
    };
    float* HnA = alloc((size_t)PP * ENC * 4); float* HnB = alloc((size_t)PP * ENC * 4);
    float* CnA = alloc((size_t)PP * ENC * 4); float* CnB = alloc((size_t)PP * ENC * 4);
    float* HeA = alloc(NN * ENC * 4); float* HeB = alloc(NN * ENC * 4);
    float* CeA = alloc(NN * ENC * 4); float* CeB = alloc(NN * ENC * 4);
    float* encB = alloc(NN * ENCIN * 4);
    float* sgrid = alloc((size_t)NN * ENC * 64 * 4);
    float* o1 = alloc((size_t)NN * C1 * 36 * 4);
    float* o2 = alloc((size_t)NN * C2 * 16 * 4);
    float* xp = alloc((size_t)NN * 512 * 4);
    float* Hz = alloc(NN * DEC * 4);
    float* CdA = alloc(NN * DEC * 4); float* CdB = alloc(NN * DEC * 4);
    float* hs = alloc((size_t)PRED_LEN * NN * DEC * 4);

    hipMemsetAsync(HnA, 0, (size_t)PP * ENC * 4, stream);
    hipMemsetAsync(CnA, 0, (size_t)PP * ENC * 4, stream);
    hipMemsetAsync(HeA, 0, NN * ENC * 4, stream);
    hipMemsetAsync(CeA, 0, NN * ENC * 4, stream);
    hipMemsetAsync(Hz, 0, NN * DEC * 4, stream);
    hipMemsetAsync(CdA, 0, NN * DEC * 4, stream);

    const size_t encSmem = ENC_SMEM_FLOATS * sizeof(float);  // ~100 KB (LDS is 320KB/WGP)
    hipFuncSetAttribute((const void*)lstm_enc_step,
                        hipFuncAttributeMaxDynamicSharedMemorySize, (int)encSmem);

    // Ego encoder: 16 timesteps over 192 state slots (ping-pong H/C).
    float *hA = HeA, *hB = HeB, *cA = CeA, *cB = CeB;
    for (int t = 0; t < OBS_LEN; ++t) {
        lstm_enc_step<<<(NN + 127) / 128, 256, encSmem, stream>>>(
            obs_traj + (size_t)t * NN * 2, obs_idxs + (size_t)t * NN,
            hA, cA, hB, cB, Wip, bip, Wih_e, Whh_e, bih_e, bhh_e, NN);
        float* tp;
        tp = hA; hA = hB; hB = tp;
        tp = cA; cA = cB; cB = tp;
    }
    float* Hs_final = hA;  // 16 swaps -> back to HeA

    // Neighbor encoder: 16 timesteps over 36864 state slots.
    float *nhA = HnA, *nhB = HnB, *ncA = CnA, *ncB = CnB;
    for (int t = 0; t < OBS_LEN; ++t) {
        lstm_enc_step<<<PP / 128, 256, encSmem, stream>>>(
            obs_ngbrs + (size_t)t * PP * 2, ngbr_idxs + (size_t)t * PP,
            nhA, ncA, nhB, ncB, Wip, bip, Wih_e, Whh_e, bih_e, bhh_e, PP);
        float* tp;
        tp = nhA; nhA = nhB; nhB = tp;
        tp = ncA; ncA = ncB; ncB = tp;
    }
    float* Hn_final = nhA;

    traj_enc_kernel<<<(NN * DYN + 255) / 256, 256, 0, stream>>>(
        Hs_final, obs_idxs + 15 * NN, Wdyn, bdyn, encB);
    soc_gather<<<(NN * ENC * 64 + 255) / 256, 256, 0, stream>>>(
        Hn_final, masks_idxs, masks, sgrid);
    conv1_k<<<(NN * C1 * 36 + 255) / 256, 256, 0, stream>>>(sgrid, Wc1, bc1, o1);
    conv2_k<<<(NN * C2 * 16 + 255) / 256, 256, 0, stream>>>(o1, Wc2, bc2, o2);
    pool_k<<<(NN * C2 * 9 + 255) / 256, 256, 0, stream>>>(o2, encB);
    xpart_k<<<(NN * 512 + 255) / 256, 256, 0, stream>>>(encB, Wih_d, bih_d, bhh_d, xp);

    // Decoder: 25 steps; hs[t] is both the output and the next step's Hprev.
    const float* hp = Hz;
    float* cp = CdA;
    float* cn = CdB;
    for (int t = 0; t < PRED_LEN; ++t) {
        float* hOut = hs + (size_t)t * NN * DEC;
        lstm_dec_step<<<24, 128, 0, stream>>>(hp, cp, hOut, cn, Whh_d, xp);
        hp = hOut;
        float* tp = cp; cp = cn; cn = tp;
    }
    out_k<<<(PRED_LEN * NN * 5 + 255) / 256, 256, 0, stream>>>(hs, Wop, bop, (float*)d_out);
}